// Mamba_64424509440084
// MI455X (gfx1250) — compile-verified
//
#include <hip/hip_runtime.h>
#include <math.h>

typedef float v2f __attribute__((ext_vector_type(2)));
typedef float v8f __attribute__((ext_vector_type(8)));

#define NB   4
#define MID  128
#define HS   32
#define SP   1024            // 32*32 spatial = D = PS
#define LSEQ 128             // scan length = channels
#define NST  16              // DSTATE
#define DTR  64              // DTRANK
#define MROWS (NB * LSEQ)    // 512 GEMM rows

// ---------------------------------------------------------------------------
// conv_in: (4,64,256,256) --8x8 s8 grouped(32)--> (4,128,32,32)
// ---------------------------------------------------------------------------
__global__ __launch_bounds__(256)
void k_conv_in(const float* __restrict__ x, const float* __restrict__ w,
               const float* __restrict__ bias, float* __restrict__ out) {
  int idx = blockIdx.x * 256 + threadIdx.x;           // 4*128*32*32
  if (idx >= NB * MID * SP) return;
  int ox = idx & 31, oy = (idx >> 5) & 31, oc = (idx >> 10) & 127, b = idx >> 17;
  int ic0 = (oc >> 2) * 2;                            // 32 groups: 4 out / 2 in
  const float* wp = w + oc * 128;                     // (oc, 2, 8, 8)
  float acc = bias[oc];
  for (int i = 0; i < 2; ++i) {
    const float* xp = x + (((size_t)b * 64 + ic0 + i) * 256 + oy * 8) * 256 + ox * 8;
    const float* wq = wp + i * 64;
#pragma unroll
    for (int ky = 0; ky < 8; ++ky)
#pragma unroll
      for (int kx = 0; kx < 8; ++kx)
        acc += xp[ky * 256 + kx] * wq[ky * 8 + kx];
  }
  out[idx] = acc;
}

// ---------------------------------------------------------------------------
// GroupNorm: 2 groups of 64 channels; one block per (b, group)
// ---------------------------------------------------------------------------
__global__ __launch_bounds__(256)
void k_groupnorm(const float* __restrict__ x, const float* __restrict__ gw,
                 const float* __restrict__ gb, float* __restrict__ out) {
  int b = blockIdx.x >> 1, g = blockIdx.x & 1;
  const float* xp = x + ((size_t)b * MID + g * 64) * SP;
  float* op = out + ((size_t)b * MID + g * 64) * SP;
  __shared__ float ssum[256], ssq[256];
  float s = 0.f, sq = 0.f;
  for (int i = threadIdx.x; i < 64 * SP; i += 256) { float v = xp[i]; s += v; sq += v * v; }
  ssum[threadIdx.x] = s; ssq[threadIdx.x] = sq;
  __syncthreads();
  for (int off = 128; off > 0; off >>= 1) {
    if ((int)threadIdx.x < off) { ssum[threadIdx.x] += ssum[threadIdx.x + off];
                                  ssq[threadIdx.x]  += ssq[threadIdx.x + off]; }
    __syncthreads();
  }
  float mean = ssum[0] * (1.0f / 65536.0f);
  float var  = ssq[0] * (1.0f / 65536.0f) - mean * mean;
  float inv  = rsqrtf(var + 1e-5f);
  for (int i = threadIdx.x; i < 64 * SP; i += 256) {
    int c = g * 64 + (i >> 10);
    op[i] = (xp[i] - mean) * inv * gw[c] + gb[c];
  }
}

// ---------------------------------------------------------------------------
// inproj: 3x3 pad1 grouped(64): (4,128,32,32) -> (4,256,32,32)
// ---------------------------------------------------------------------------
__global__ __launch_bounds__(256)
void k_inproj(const float* __restrict__ x, const float* __restrict__ w,
              float* __restrict__ out) {
  int idx = blockIdx.x * 256 + threadIdx.x;           // 4*256*1024
  if (idx >= NB * 2 * MID * SP) return;
  int sx = idx & 31, sy = (idx >> 5) & 31, oc = (idx >> 10) & 255, b = idx >> 18;
  int ic0 = (oc >> 2) * 2;                            // 64 groups: 4 out / 2 in
  const float* wp = w + oc * 18;                      // (oc, 2, 3, 3)
  float acc = 0.f;
  for (int i = 0; i < 2; ++i) {
    const float* xp = x + ((size_t)b * MID + ic0 + i) * SP;
    const float* wq = wp + i * 9;
#pragma unroll
    for (int ky = -1; ky <= 1; ++ky) {
      int yy = sy + ky; if (yy < 0 || yy > 31) continue;
#pragma unroll
      for (int kx = -1; kx <= 1; ++kx) {
        int xx = sx + kx; if (xx < 0 || xx > 31) continue;
        acc += xp[yy * 32 + xx] * wq[(ky + 1) * 3 + (kx + 1)];
      }
    }
  }
  out[idx] = acc;
}

// ---------------------------------------------------------------------------
// 1x1 grouped(64) conv + bias + SiLU : xr[:, :128] -> u  (b, L=128, D=1024)
// ---------------------------------------------------------------------------
__global__ __launch_bounds__(256)
void k_dwproj_silu(const float* __restrict__ xr, const float* __restrict__ w,
                   const float* __restrict__ bias, float* __restrict__ u) {
  int idx = blockIdx.x * 256 + threadIdx.x;           // 4*128*1024
  if (idx >= NB * MID * SP) return;
  int s = idx & 1023, oc = (idx >> 10) & 127, b = idx >> 17;
  int ic0 = oc & ~1;                                  // 64 groups: 2 out / 2 in
  const float* xp = xr + (size_t)b * 2 * MID * SP;
  float v = xp[(size_t)ic0 * SP + s] * w[oc * 2]
          + xp[(size_t)(ic0 + 1) * SP + s] * w[oc * 2 + 1] + bias[oc];
  u[idx] = v / (1.0f + expf(-v));                     // silu
}

// ---------------------------------------------------------------------------
// WMMA f32 GEMM: out[m][n] = act( sum_k A[m,k] * W[n,k] + bias[n] )
// A: (M, lda) row-major.  W: (N, ldw) row-major (i.e. B = W^T).
// One wave per 16 x (16*NT) output strip.  Per K-step: batch ALL loads
// (1 A tile + NT B tiles) first so they clause and waits are incremental,
// then issue NT V_WMMA_F32_16X16X4_F32 against the shared A tile.
// ACT: 0 = none, 1 = softplus.
// ---------------------------------------------------------------------------
template <int ACT, int NT>
__global__ __launch_bounds__(32)
void k_gemm_wmma_bt(const float* __restrict__ A, int lda,
                    const float* __restrict__ W, int ldw,
                    const float* __restrict__ bias,
                    float* __restrict__ out, int ldo, int K) {
  const int lane = threadIdx.x & 31;
  const int half = lane >> 4;                         // K-pair selector
  const int l16  = lane & 15;
  const int m0 = blockIdx.x * 16;
  const int n0 = blockIdx.y * (16 * NT);
  // A 16x4 layout: lane row = m0+l16; VGPR{0,1} hold K = 2*half + {0,1}
  const float* arow = A + (size_t)(m0 + l16) * lda + 2 * half;
  // B 4x16 layout: lane col = n+l16; VGPR{0,1} hold K = 2*half + {0,1}
  const float* wrow[NT];
#pragma unroll
  for (int t = 0; t < NT; ++t)
    wrow[t] = W + (size_t)(n0 + 16 * t + l16) * ldw + 2 * half;

  v8f c[NT];
#pragma unroll
  for (int t = 0; t < NT; ++t)
    c[t] = (v8f){0.f, 0.f, 0.f, 0.f, 0.f, 0.f, 0.f, 0.f};

  for (int k = 0; k < K; k += 4) {
    if ((k & 63) == 0 && k + 64 < K)                  // gfx1250 global_prefetch_b8
      __builtin_prefetch(arow + k + 64, 0, 1);
    // --- batch all loads for this K-step ---
    v2f a, b[NT];
    a = *(const v2f*)(arow + k);
#pragma unroll
    for (int t = 0; t < NT; ++t)
      b[t] = *(const v2f*)(wrow[t] + k);
    // --- then the WMMA group (A reused NT times) ---
#pragma unroll
    for (int t = 0; t < NT; ++t)
      c[t] = __builtin_amdgcn_wmma_f32_16x16x4_f32(
          /*neg_a=*/false, a, /*neg_b=*/false, b[t],
          /*c_mod=*/(short)0, c[t], /*reuse_a=*/false, /*reuse_b=*/false);
  }
#pragma unroll
  for (int t = 0; t < NT; ++t) {
    float bn = bias ? bias[n0 + 16 * t + l16] : 0.0f;
#pragma unroll
    for (int v = 0; v < 8; ++v) {
      int row = m0 + v + 8 * half;                    // C/D layout
      float val = c[t][v] + bn;
      if (ACT == 1) val = (val > 20.0f) ? val : log1pf(expf(val));  // softplus
      out[(size_t)row * ldo + n0 + 16 * t + l16] = val;
    }
  }
}

// ---------------------------------------------------------------------------
// Selective scan fused with output gate + residual:
//   h_out[b,l,d] = (scan_y[b,l,d]) * silu(res[b,l,d]) + h_in[b,l,d]
// One thread per (b, d); state s[16] in VGPRs; B/C staged in LDS (16 KB).
// ---------------------------------------------------------------------------
__global__ __launch_bounds__(256)
void k_scan_gate(const float* __restrict__ u, const float* __restrict__ delta,
                 const float* __restrict__ xdbl, const float* __restrict__ alog,
                 const float* __restrict__ dd, const float* __restrict__ xr,
                 const float* __restrict__ hin, float* __restrict__ hout) {
  __shared__ float bc[LSEQ * 32];
  int t = blockIdx.x * 256 + threadIdx.x;             // 4096 scans
  int b = t >> 10, d = t & 1023;
  for (int i = threadIdx.x; i < LSEQ * 32; i += 256) {
    int l = i >> 5, j = i & 31;
    bc[i] = xdbl[(size_t)(b * LSEQ + l) * 96 + 64 + j];   // Bm | Cm
  }
  __syncthreads();
  float An[NST];
#pragma unroll
  for (int n = 0; n < NST; ++n) An[n] = -expf(alog[d * NST + n]);
  float Dd = dd[d];
  float s[NST];
#pragma unroll
  for (int n = 0; n < NST; ++n) s[n] = 0.0f;
  for (int l = 0; l < LSEQ; ++l) {
    size_t off = (size_t)(b * LSEQ + l) * SP + d;
    float dl = delta[off], uu = u[off];
    float du = dl * uu;
    const float* Bp = &bc[l * 32];
    float acc = 0.0f;
#pragma unroll
    for (int n = 0; n < NST; ++n) {
      s[n] = expf(dl * An[n]) * s[n] + du * Bp[n];
      acc += s[n] * Bp[16 + n];
    }
    float y = acc + uu * Dd;
    float r = xr[((size_t)(b * 2 * MID + MID + l)) * SP + d];   // res channel
    float g = r / (1.0f + expf(-r));                            // silu
    hout[off] = y * g + hin[off];
  }
}

// ---------------------------------------------------------------------------
// conv_out: 1x1 grouped(32): 4 in / 4 out per group
// ---------------------------------------------------------------------------
__global__ __launch_bounds__(256)
void k_conv_out(const float* __restrict__ h, const float* __restrict__ w,
                const float* __restrict__ bias, float* __restrict__ out) {
  int idx = blockIdx.x * 256 + threadIdx.x;           // 4*128*1024
  if (idx >= NB * MID * SP) return;
  int s = idx & 1023, oc = (idx >> 10) & 127, b = idx >> 17;
  int ic0 = (oc >> 2) * 4;
  const float* hp = h + ((size_t)b * MID + ic0) * SP + s;
  const float* wp = w + oc * 4;
  float acc = bias[oc];
#pragma unroll
  for (int i = 0; i < 4; ++i) acc += hp[(size_t)i * SP] * wp[i];
  out[idx] = acc;
}

// ---------------------------------------------------------------------------
// Bilinear x8 upsample, half-pixel centers, edge clamp: (4,128,32,32)->(..,256,256)
// ---------------------------------------------------------------------------
__global__ __launch_bounds__(256)
void k_resize8(const float* __restrict__ in, float* __restrict__ out) {
  size_t idx = (size_t)blockIdx.x * 256 + threadIdx.x;  // 4*128*256*256
  if (idx >= (size_t)NB * MID * 256 * 256) return;
  int X = idx & 255, Y = (idx >> 8) & 255;
  int c = (idx >> 16) & 127; int b = (int)(idx >> 23);
  float sy = (Y + 0.5f) * 0.125f - 0.5f;
  float sx = (X + 0.5f) * 0.125f - 0.5f;
  int y0 = (int)floorf(sy), x0 = (int)floorf(sx);
  float fy = sy - y0, fx = sx - x0;
  int y0c = min(max(y0, 0), 31), y1c = min(max(y0 + 1, 0), 31);
  int x0c = min(max(x0, 0), 31), x1c = min(max(x0 + 1, 0), 31);
  const float* p = in + ((size_t)b * MID + c) * SP;
  float v00 = p[y0c * 32 + x0c], v01 = p[y0c * 32 + x1c];
  float v10 = p[y1c * 32 + x0c], v11 = p[y1c * 32 + x1c];
  float v0 = v00 + (v01 - v00) * fx;
  float v1 = v10 + (v11 - v10) * fx;
  out[idx] = v0 + (v1 - v0) * fy;
}

// ---------------------------------------------------------------------------
extern "C" void kernel_launch(void* const* d_in, const int* in_sizes, int n_in,
                              void* d_out, int out_size, void* d_ws, size_t ws_size,
                              hipStream_t stream) {
  (void)in_sizes; (void)n_in; (void)out_size; (void)ws_size;
  const float* x        = (const float*)d_in[0];
  const float* w_in     = (const float*)d_in[1];
  const float* b_in     = (const float*)d_in[2];
  // per-block params, block 0 at indices 3..12, block 1 at 13..22
  const float* w_out    = (const float*)d_in[23];
  const float* b_out    = (const float*)d_in[24];

  const size_t S = (size_t)NB * MID * SP;             // 524288 floats
  float* ws    = (float*)d_ws;
  float* hA    = ws;                                  // stream buffer A
  float* hB    = ws + S;                              // stream buffer B
  float* gn    = ws + 2 * S;                          // also reused for conv_out
  float* xr    = ws + 3 * S;                          // (4,256,32,32) = 2S
  float* u     = ws + 5 * S;
  float* delta = ws + 6 * S;
  float* xdbl  = ws + 7 * S;                          // (512, 96)

  // Stage 0: input conv
  k_conv_in<<<(NB * MID * SP + 255) / 256, 256, 0, stream>>>(x, w_in, b_in, hA);

  // Two mamba blocks
  float* hin = hA; float* hout = hB;
  for (int blk = 0; blk < 2; ++blk) {
    int p = 3 + blk * 10;
    const float* gn_g    = (const float*)d_in[p + 0];
    const float* gn_b    = (const float*)d_in[p + 1];
    const float* inproj_w= (const float*)d_in[p + 2];
    const float* conv_w  = (const float*)d_in[p + 3];
    const float* conv_b  = (const float*)d_in[p + 4];
    const float* xproj_w = (const float*)d_in[p + 5];
    const float* dtp_w   = (const float*)d_in[p + 6];
    const float* dtp_b   = (const float*)d_in[p + 7];
    const float* alog    = (const float*)d_in[p + 8];
    const float* dd      = (const float*)d_in[p + 9];

    k_groupnorm<<<NB * 2, 256, 0, stream>>>(hin, gn_g, gn_b, gn);
    k_inproj<<<(NB * 2 * MID * SP + 255) / 256, 256, 0, stream>>>(gn, inproj_w, xr);
    k_dwproj_silu<<<(NB * MID * SP + 255) / 256, 256, 0, stream>>>(xr, conv_w, conv_b, u);
    // x_dbl = u (512x1024) @ xproj_w^T (1024x96): 16x48 strip per wave (NT=3)
    k_gemm_wmma_bt<0, 3><<<dim3(MROWS / 16, 96 / 48), 32, 0, stream>>>(
        u, SP, xproj_w, SP, nullptr, xdbl, 96, SP);
    // delta = softplus(x_dbl[:, :64] @ dtp_w^T + dtp_b): 16x64 strip (NT=4)
    k_gemm_wmma_bt<1, 4><<<dim3(MROWS / 16, SP / 64), 32, 0, stream>>>(
        xdbl, 96, dtp_w, DTR, dtp_b, delta, SP, DTR);
    // scan + gate + residual fused
    k_scan_gate<<<NB * SP / 256, 256, 0, stream>>>(
        u, delta, xdbl, alog, dd, xr, hin, hout);

    float* t = hin; hin = hout; hout = t;             // hin now holds block output
  }

  // Output head: 1x1 grouped conv into gn buffer, then x8 bilinear into d_out
  k_conv_out<<<(NB * MID * SP + 255) / 256, 256, 0, stream>>>(hin, w_out, b_out, gn);
  k_resize8<<<(unsigned)(((size_t)NB * MID * 256 * 256 + 255) / 256), 256, 0, stream>>>(
      gn, (float*)d_out);
}